// VariableLengthTransferFunctionLoss_87153476370968
// MI455X (gfx1250) — compile-verified
//
#include <hip/hip_runtime.h>
#include <math.h>

typedef float v2f __attribute__((ext_vector_type(2)));
typedef float v8f __attribute__((ext_vector_type(8)));

#define BIGF 1e30f

// ---------- wave32 reduction helpers ----------
__device__ __forceinline__ float wave_sum_f(float v) {
#pragma unroll
    for (int m = 16; m >= 1; m >>= 1) v += __shfl_xor(v, m, 32);
    return v;
}

// ---------- WMMA 16x16x4 f32, C = inline 0 ----------
__device__ __forceinline__ v8f wmma4(v2f a, v2f b) {
    v8f c = {};
    // 8 args: (neg_a, A, neg_b, B, c_mod, C, reuse_a, reuse_b)
    return __builtin_amdgcn_wmma_f32_16x16x4_f32(false, a, false, b, (short)0, c,
                                                 false, false);
}

// One chamfer direction A->B.
// Computes, per lane, min over valid cols j<nB of ( |b_j|^2 - 2*a_i.b_j )
// for row i(lane) = 16*((lane>>3)&1) + (lane&7) + 8*(lane>=16)  (all 32 rows,
// one per lane). The |b_j|^2 term is folded into the WMMA via the K=2 slot:
//   A[i, 0..3] = (-2*a.x, -2*a.y, 1, 0)   B[0..3, j] = (b.x, b.y, |b_j|^2, 0)
// a_lo/a_hi = A points at index (lane&15) and 16+(lane&15); same for b.
// bn0/bn1   = |b|^2 at those indices.
__device__ __forceinline__ float rowmin_dir(
    float2 a_lo, float2 a_hi, float2 b_lo, float2 b_hi,
    float bn0, float bn1, int nB, int lane)
{
    const bool lo = lane < 16;
    // A tiles: K=0,1 (lanes 0-15) = -2*point; K=2 (V0, lanes 16-31) = 1, K=3 = 0
    v2f A0, A1, B0, B1;
    A0.x = lo ? -2.0f * a_lo.x : 1.0f;  A0.y = lo ? -2.0f * a_lo.y : 0.0f;
    A1.x = lo ? -2.0f * a_hi.x : 1.0f;  A1.y = lo ? -2.0f * a_hi.y : 0.0f;
    // B tiles: K=0,1 rows (lanes 0-15) = point; K=2 row (V0, lanes 16-31) = |b_j|^2
    B0.x = lo ? b_lo.x : bn0;           B0.y = lo ? b_lo.y : 0.0f;
    B1.x = lo ? b_hi.x : bn1;           B1.y = lo ? b_hi.y : 0.0f;

    // d[mt][nt][r] = |b_j|^2 - 2 a_i.b_j at i = mt*16 + r + 8*(lane>=16),
    //                                      j = nt*16 + (lane&15)
    v8f d00 = wmma4(A0, B0);
    v8f d01 = wmma4(A0, B1);
    v8f d10 = wmma4(A1, B0);
    v8f d11 = wmma4(A1, B1);

    const bool v0 = (lane & 15) < nB;         // j = lane&15 valid
    const bool v1 = (16 + (lane & 15)) < nB;  // j = 16+(lane&15) valid

    // t[q], q = mt*8 + r : per-lane partial min over the 2 columns it holds
    float t[16];
#pragma unroll
    for (int r = 0; r < 8; ++r) {
        t[r]     = fminf(v0 ? d00[r] : BIGF, v1 ? d01[r] : BIGF);
        t[8 + r] = fminf(v0 ? d10[r] : BIGF, v1 ? d11[r] : BIGF);
    }

    // Multi-row butterfly: 16 regs -> 1 reg in 15 shuffles; shuffle masks 1,2,4,8
    // stay within 16-lane halves. Afterwards lane l tracks register q = l&15
    // fully reduced over its half's 16 lanes.
#pragma unroll
    for (int s = 0; s < 4; ++s) {
        const bool bit = (lane >> s) & 1;
#pragma unroll
        for (int k = 0; k < (8 >> s); ++k) {
            float a = t[2 * k];
            float b = t[2 * k + 1];
            float mine  = bit ? b : a;
            float other = bit ? a : b;
            t[k] = fminf(mine, __shfl_xor(other, 1 << s, 32));
        }
    }
    return t[0];
}

// Per-lane partial of the variable-length chamfer for one batch row
// (caller wave-sums). P,T: 32 float2 points each.
__device__ __forceinline__ float chamfer_partial(const float2* __restrict__ P,
                                                 const float2* __restrict__ T,
                                                 int nP, int nT, int lane)
{
    const int li = lane & 15;
    float2 p_lo = P[li], p_hi = P[16 + li];
    float2 t_lo = T[li], t_hi = T[16 + li];
    float pn0 = fmaf(p_lo.x, p_lo.x, p_lo.y * p_lo.y);
    float pn1 = fmaf(p_hi.x, p_hi.x, p_hi.y * p_hi.y);
    float tn0 = fmaf(t_lo.x, t_lo.x, t_lo.y * t_lo.y);
    float tn1 = fmaf(t_hi.x, t_hi.x, t_hi.y * t_hi.y);
    // lane l holds |point l|^2; mask to valid indices
    float pnm = (lane < nP) ? ((lane < 16) ? pn0 : pn1) : 0.0f;
    float tnm = (lane < nT) ? ((lane < 16) ? tn0 : tn1) : 0.0f;

    float rm_p2t = rowmin_dir(p_lo, p_hi, t_lo, t_hi, tn0, tn1, nT, lane);
    float rm_t2p = rowmin_dir(t_lo, t_hi, p_lo, p_hi, pn0, pn1, nP, lane);

    // row index tracked by this lane after the butterfly
    const int irow = 16 * ((lane >> 3) & 1) + (lane & 7) + ((lane >= 16) ? 8 : 0);
    float rmp = (irow < nP) ? rm_p2t : 0.0f;  // p2t rows are pred indices
    float rmt = (irow < nT) ? rm_t2p : 0.0f;  // t2p rows are target indices

    float invP = 1.0f / fmaxf((float)nP, 1.0f);
    float invT = 1.0f / fmaxf((float)nT, 1.0f);
    float normal = (pnm + rmp) * invP + (tnm + rmt) * invT;

    const bool p0 = (nP == 0), t0 = (nT == 0);
    // wave-uniform select of the per-lane partial
    return (p0 && t0) ? 0.0f : (p0 ? tnm : (t0 ? pnm : normal));
}

// Cross-entropy NLL + argmax (first occurrence) over 33-class logit row.
__device__ __forceinline__ float ce_and_argmax(const float* __restrict__ row,
                                               int tgt, int lane, int* n_out)
{
    float x = row[lane];
    float x32 = row[32];
    float v = x;
    int idx = lane;
    bool take32 = (lane == 0) && (x32 > x);   // tie -> keep index 0
    v = take32 ? x32 : v;
    idx = take32 ? 32 : idx;
#pragma unroll
    for (int m = 16; m >= 1; m >>= 1) {
        float ov = __shfl_xor(v, m, 32);
        int oi = __shfl_xor(idx, m, 32);
        bool t = (ov > v) || ((ov == v) && (oi < idx));
        v = t ? ov : v;
        idx = t ? oi : idx;
    }
    float s = expf(x - v) + ((lane == 0) ? expf(x32 - v) : 0.0f);
    s = wave_sum_f(s);
    *n_out = idx;
    return v + logf(s) - row[tgt];   // logsumexp - logit[target]
}

__global__ __launch_bounds__(256) void tf_loss_kernel(
    const float* __restrict__ pole_logits, const float* __restrict__ zero_logits,
    const float* __restrict__ poles, const float* __restrict__ zeros_,
    const float* __restrict__ tpoles, const float* __restrict__ tzeros,
    const int* __restrict__ tnp, const int* __restrict__ tnz,
    float* __restrict__ partials)
{
    const int lane = threadIdx.x & 31;
    const int wave = threadIdx.x >> 5;
    const int b = blockIdx.x * 8 + wave;

    const int npt = tnp[b];
    const int nzt = tnz[b];

    int npp, nzp;
    float nll_p = ce_and_argmax(pole_logits + (size_t)b * 33, npt, lane, &npp);
    float nll_z = ce_and_argmax(zero_logits + (size_t)b * 33, nzt, lane, &nzp);

    float ch_pl = chamfer_partial((const float2*)poles  + (size_t)b * 32,
                                  (const float2*)tpoles + (size_t)b * 32, npp, npt, lane);
    float ch_zl = chamfer_partial((const float2*)zeros_ + (size_t)b * 32,
                                  (const float2*)tzeros + (size_t)b * 32, nzp, nzt, lane);
    // single wave reduction for both chamfer terms
    float ch_sum = wave_sum_f(ch_pl + ch_zl);

    float contrib = 5.0f * nll_p + 5.0f * nll_z + ch_sum;

    __shared__ float sacc[8];
    if (lane == 0) sacc[wave] = contrib;
    __syncthreads();
    if (threadIdx.x == 0) {
        float t = 0.0f;
#pragma unroll
        for (int i = 0; i < 8; ++i) t += sacc[i];
        partials[blockIdx.x] = t;
    }
}

__global__ __launch_bounds__(256) void reduce_kernel(
    const float* __restrict__ partials, int n, float inv_b, float* __restrict__ out)
{
    __shared__ float s[256];
    float a = 0.0f;
    for (int i = threadIdx.x; i < n; i += 256) a += partials[i];
    s[threadIdx.x] = a;
    __syncthreads();
    for (int st = 128; st > 0; st >>= 1) {
        if (threadIdx.x < st) s[threadIdx.x] += s[threadIdx.x + st];
        __syncthreads();
    }
    if (threadIdx.x == 0) out[0] = s[0] * inv_b;
}

extern "C" void kernel_launch(void* const* d_in, const int* in_sizes, int n_in,
                              void* d_out, int out_size, void* d_ws, size_t ws_size,
                              hipStream_t stream) {
    const float* pole_logits = (const float*)d_in[0];
    const float* zero_logits = (const float*)d_in[1];
    const float* poles       = (const float*)d_in[2];
    const float* zeros_      = (const float*)d_in[3];
    const float* tpoles      = (const float*)d_in[4];
    const float* tzeros      = (const float*)d_in[5];
    const int*   tnp         = (const int*)d_in[6];
    const int*   tnz         = (const int*)d_in[7];

    const int B = in_sizes[6];            // 32768
    const int blocks = B / 8;             // 8 waves (batch rows) per 256-thread block
    float* partials = (float*)d_ws;

    tf_loss_kernel<<<blocks, 256, 0, stream>>>(pole_logits, zero_logits, poles, zeros_,
                                               tpoles, tzeros, tnp, tnz, partials);
    reduce_kernel<<<1, 256, 0, stream>>>(partials, blocks, 1.0f / (float)B,
                                         (float*)d_out);
}